// CARAFENAIVE_40389872452090
// MI455X (gfx1250) — compile-verified
//
#include <hip/hip_runtime.h>

// CARAFE (N=2, C=256, H=W=100, K=5, G=1, S=2) as a banded fp32 WMMA GEMM.
//
// Per (n, h, ow-tile of 16): out[c, 2h+r, ow0+j] = sum_{dy,ws} A[c][dy][ws] * B[r][dy][ws][j]
//   A[c][dy][ws]    = feat[n, c, h+dy-2, ow0/2-2+ws]      (zero-padded, ws in [0,12))
//   B[r][dy][ws][j] = mask[n, dy*5+(ws-j/2), 2h+r, ow0+j] if 0<=ws-j/2<5 else 0
// K = 5*12 = 60 -> 15 steps of V_WMMA_F32_16X16X4_F32 per output row (fp32-exact).

typedef __attribute__((ext_vector_type(2))) float v2f;
typedef __attribute__((ext_vector_type(8))) float v8f;

#define NN 2
#define CC 256
#define HH 100
#define WW 100
#define OHH 200
#define OWW 200

__global__ __launch_bounds__(512)
void CARAFE_wmma_f32_kernel(const float* __restrict__ feat,
                            const float* __restrict__ mask,
                            float* __restrict__ out) {
    // LDS B: [row r(2)][dy(5)][kpair(6)][j(16)][parity(2)] -> fragment load is one b64
    __shared__ float Bs[2 * 5 * 6 * 16 * 2]; // 1920 floats = 7.5 KB

    const int bx  = blockIdx.x;      // ow tile (16 wide), 13 tiles
    const int h   = blockIdx.y;      // source row, covers oh = 2h, 2h+1
    const int n   = blockIdx.z;
    const int ow0   = bx * 16;
    const int wbase = ow0 / 2 - 2;   // leftmost source column for this tile
    const int tid = threadIdx.x;

    // ---- cooperative banded-B build in LDS (masks read once per WG) ----
    for (int e = tid; e < 1920; e += 512) {
        const int par = e & 1;
        const int j   = (e >> 1) & 15;
        const int kp  = (e >> 5) % 6;
        const int dy  = ((e >> 5) / 6) % 5;
        const int r   = e / 960;
        const int ws  = kp * 2 + par;          // k within this dy, 0..11
        const int dx  = ws - (j >> 1);
        const int ow  = ow0 + j;
        float v = 0.0f;
        if (dx >= 0 && dx < 5 && ow < OWW) {
            const int oh = 2 * h + r;
            v = mask[((n * 25 + (dy * 5 + dx)) * OHH + oh) * OWW + ow];
        }
        Bs[e] = v;
    }
    __syncthreads();

    const int wave = tid >> 5;       // 16 waves -> 16 channel tiles
    const int lane = tid & 31;
    const int half = lane >> 4;      // 0: lanes 0-15, 1: lanes 16-31
    const int l16  = lane & 15;
    const int c0   = wave * 16;
    const int ch   = c0 + l16;

    // ---- A fragments: lane layout for 32-bit A 16x4 (ISA 7.12.2):
    // lanes 0-15 hold K=ks,ks+1 ; lanes 16-31 hold K=ks+2,ks+3 ; M = lane%16
    v2f a[5][3];
    #pragma unroll
    for (int dy = 0; dy < 5; ++dy) {
        const int row = h + dy - 2;
        const bool rowok = (row >= 0) && (row < HH);
        const long base = (((long)(n * CC + ch)) * HH + row) * WW;
        #pragma unroll
        for (int s = 0; s < 3; ++s) {
            const int w0 = wbase + 4 * s + 2 * half;
            v2f av; av.x = 0.0f; av.y = 0.0f;
            if (rowok) {
                if (w0 >= 0 && w0 < WW)         av.x = feat[base + w0];
                if (w0 + 1 >= 0 && w0 + 1 < WW) av.y = feat[base + w0 + 1];
            }
            a[dy][s] = av;
        }
    }

    // ---- B fragments from LDS (mirrored K striping: k = ks + 2*half + v) ----
    v2f b[2][5][3];
    #pragma unroll
    for (int r = 0; r < 2; ++r)
        #pragma unroll
        for (int dy = 0; dy < 5; ++dy)
            #pragma unroll
            for (int s = 0; s < 3; ++s) {
                const int idx = ((((r * 5 + dy) * 6 + (2 * s + half)) * 16 + l16) * 2);
                b[r][dy][s] = *(const v2f*)&Bs[idx];
            }

    // ---- 30 x V_WMMA_F32_16X16X4_F32, two independent accumulator chains ----
    v8f acc0 = {};
    v8f acc1 = {};
    #pragma unroll
    for (int dy = 0; dy < 5; ++dy)
        #pragma unroll
        for (int s = 0; s < 3; ++s) {
            acc0 = __builtin_amdgcn_wmma_f32_16x16x4_f32(
                false, a[dy][s], false, b[0][dy][s], (short)0, acc0, false, false);
            acc1 = __builtin_amdgcn_wmma_f32_16x16x4_f32(
                false, a[dy][s], false, b[1][dy][s], (short)0, acc1, false, false);
        }

    // ---- store: C/D layout VGPR v -> M = v + 8*half, N = lane%16 ----
    const int j = l16;
    if (ow0 + j < OWW) {
        #pragma unroll
        for (int v = 0; v < 8; ++v) {
            const int chv = c0 + v + 8 * half;
            out[(((long)(n * CC + chv)) * OHH + (2 * h + 0)) * OWW + ow0 + j] = acc0[v];
            out[(((long)(n * CC + chv)) * OHH + (2 * h + 1)) * OWW + ow0 + j] = acc1[v];
        }
    }
}

extern "C" void kernel_launch(void* const* d_in, const int* in_sizes, int n_in,
                              void* d_out, int out_size, void* d_ws, size_t ws_size,
                              hipStream_t stream) {
    const float* feat = (const float*)d_in[0]; // [2,256,100,100] f32
    const float* mask = (const float*)d_in[1]; // [2,25,200,200] f32
    float* out = (float*)d_out;                // [2,256,200,200] f32
    (void)in_sizes; (void)n_in; (void)out_size; (void)d_ws; (void)ws_size;

    dim3 grid((OWW + 15) / 16, HH, NN);        // 13 x 100 x 2
    dim3 block(512);                           // 16 waves; 1 channel-tile each
    CARAFE_wmma_f32_kernel<<<grid, block, 0, stream>>>(feat, mask, out);
}